// ST_BIFNeuron_34376918237347
// MI455X (gfx1250) — compile-verified
//
#include <hip/hip_runtime.h>
#include <hip/hip_bf16.h>
#include <math.h>

typedef __attribute__((ext_vector_type(2))) float v2f;
typedef __attribute__((ext_vector_type(4))) float v4f;
typedef __attribute__((ext_vector_type(8))) float v8f;

#define ST_LEVEL 16
#define RED_BLOCKS 2048
#define SCAN_BLOCK 256

// ---------------------------------------------------------------------------
// Pass 1: time-sum over the 16 time rows via V_WMMA_F32_16X16X4_F32.
// A = ones(16x4) so D[i][j] = sum_k B[k][j]; chaining 4 WMMAs over K covers
// all 16 time steps in exact fp32.
//
// Column grouping into WMMA tiles is arbitrary for this reduction, so lane n
// of chain j owns column m0 + 4n + j: each lane then loads a contiguous
// float4 (global_load_b128) per time row and feeds 4 independent accumulator
// chains from its components. 64 columns / 8 b128 loads / 16 WMMAs per group.
// Every D row is identical and the two lane halves duplicate the same
// columns, so sum(|acc[0]|) * 0.5 over the wave is the group's contribution.
// ---------------------------------------------------------------------------
__global__ void stbif_reduce_wmma(const float* __restrict__ x,
                                  float* __restrict__ partials,
                                  long long M)
{
    const int lane          = threadIdx.x & 31;
    const int wave          = threadIdx.x >> 5;
    const int wavesPerBlock = blockDim.x >> 5;
    const long long gwave   = (long long)blockIdx.x * wavesPerBlock + wave;
    const long long nwaves  = (long long)gridDim.x * wavesPerBlock;
    const long long ngroups = M >> 6;            // 64 columns per group

    v2f a; a.x = 1.0f; a.y = 1.0f;               // A = ones(16x4), 2 VGPRs

    const int half = lane >> 4;                  // 0: K rows {4k,4k+1}, 1: {4k+2,4k+3}
    const int l16  = lane & 15;                  // lane's column slot

    float partial = 0.0f;
    for (long long g = gwave; g < ngroups; g += nwaves) {
        const long long m0 = g << 6;
        const float* colbase = x + m0 + 4 * l16; // lane's 4 adjacent columns

        v8f acc0 = {}, acc1 = {}, acc2 = {}, acc3 = {};
#pragma unroll
        for (int k = 0; k < 4; ++k) {
            const float* p = colbase + (long long)(4 * k + 2 * half) * M;
            const v4f r0 = *(const v4f*)p;        // time row r
            const v4f r1 = *(const v4f*)(p + M);  // time row r+1
            v2f b;
            b.x = r0[0]; b.y = r1[0];
            acc0 = __builtin_amdgcn_wmma_f32_16x16x4_f32(false, a, false, b, (short)0, acc0, false, false);
            b.x = r0[1]; b.y = r1[1];
            acc1 = __builtin_amdgcn_wmma_f32_16x16x4_f32(false, a, false, b, (short)0, acc1, false, false);
            b.x = r0[2]; b.y = r1[2];
            acc2 = __builtin_amdgcn_wmma_f32_16x16x4_f32(false, a, false, b, (short)0, acc2, false, false);
            b.x = r0[3]; b.y = r1[3];
            acc3 = __builtin_amdgcn_wmma_f32_16x16x4_f32(false, a, false, b, (short)0, acc3, false, false);
        }
        // all D rows identical; lane halves duplicate the same columns -> *0.5
        partial += 0.5f * (fabsf(acc0[0]) + fabsf(acc1[0]) +
                           fabsf(acc2[0]) + fabsf(acc3[0]));
    }

    // wave32 reduction
    for (int off = 16; off > 0; off >>= 1)
        partial += __shfl_down(partial, off, 32);

    __shared__ float sdata[32];
    if (lane == 0) sdata[wave] = partial;
    __syncthreads();
    if (threadIdx.x == 0) {
        float s = 0.0f;
        for (int w = 0; w < wavesPerBlock; ++w) s += sdata[w];
        partials[blockIdx.x] = s;
    }
}

// ---------------------------------------------------------------------------
// Deterministic finalize: fixed-order sum of the per-block partials, then
// v_th = mean * 2 / sqrt(15).
// ---------------------------------------------------------------------------
__global__ void stbif_finalize(const float* __restrict__ partials, int n,
                               float* __restrict__ vth_out, long long M)
{
    __shared__ float s[256];
    float acc = 0.0f;
    for (int i = threadIdx.x; i < n; i += blockDim.x) acc += partials[i];
    s[threadIdx.x] = acc;
    __syncthreads();
    for (int off = 128; off > 0; off >>= 1) {
        if ((int)threadIdx.x < off) s[threadIdx.x] += s[threadIdx.x + off];
        __syncthreads();
    }
    if (threadIdx.x == 0) {
        float mean = s[0] / (float)M;
        vth_out[0] = mean * 2.0f / sqrtf(15.0f);
    }
}

// ---------------------------------------------------------------------------
// Pass 2: per-column 16-step integrate-and-fire scan, fused with the s_scale
// multiply. Each thread carries 4 adjacent columns so all traffic is b128.
// Non-temporal loads (last use of input) and non-temporal stores (output is
// never re-read) keep L2 available for pass-1 residual lines.
// ---------------------------------------------------------------------------
__global__ void stbif_scan(const float* __restrict__ x,
                           float* __restrict__ out,
                           const float* __restrict__ vth_p,
                           long long M)
{
    const float vth = vth_p[0];
    const long long nq = M >> 2;                 // float4 columns
    const long long stride = (long long)gridDim.x * blockDim.x;
    const v4f* __restrict__ xq = (const v4f*)x;
    v4f* __restrict__ oq = (v4f*)out;

    for (long long q = (long long)blockIdx.x * blockDim.x + threadIdx.x;
         q < nq; q += stride) {
        float v[4];
        int   T[4];
#pragma unroll
        for (int j = 0; j < 4; ++j) { v[j] = 0.5f * vth; T[j] = 0; }

#pragma unroll
        for (int t = 0; t < ST_LEVEL; ++t) {
            const long long idx = (long long)t * nq + q;
            const v4f xv = __builtin_nontemporal_load(&xq[idx]);
            v4f ov;
#pragma unroll
            for (int j = 0; j < 4; ++j) {
                v[j] += xv[j];
                const bool pos = ((v[j] - vth) >= 0.0f) && (T[j] < (ST_LEVEL - 1));
                const bool neg = (v[j] < 0.0f) && (T[j] > 0);
                const float sp = neg ? -1.0f : (pos ? 1.0f : 0.0f);
                v[j] -= vth * sp;
                T[j] += (int)sp;
                ov[j] = sp * vth;
            }
            __builtin_nontemporal_store(ov, &oq[idx]);
        }
    }
}

extern "C" void kernel_launch(void* const* d_in, const int* in_sizes, int n_in,
                              void* d_out, int out_size, void* d_ws, size_t ws_size,
                              hipStream_t stream)
{
    const float* x   = (const float*)d_in[0];
    float*       out = (float*)d_out;

    const long long N = (long long)in_sizes[0];     // 67,108,864
    const long long M = N / ST_LEVEL;               // 4,194,304 columns

    float* ws       = (float*)d_ws;
    float* vth      = ws;                           // ws[0]
    float* partials = ws + 8;                       // RED_BLOCKS floats

    // Pass 1: WMMA time-sum + |.| reduction -> per-block partials
    stbif_reduce_wmma<<<RED_BLOCKS, 256, 0, stream>>>(x, partials, M);

    // Finalize v_th (deterministic single-block tree)
    stbif_finalize<<<1, 256, 0, stream>>>(partials, RED_BLOCKS, vth, M);

    // Pass 2: fused scan + scale, 4 columns per thread (b128 traffic)
    int scan_blocks = (int)(((M >> 2) + SCAN_BLOCK - 1) / SCAN_BLOCK); // 4096
    stbif_scan<<<scan_blocks, SCAN_BLOCK, 0, stream>>>(x, out, vth, M);
}